// FrameOTAMDistanceNetwork_63591285785137
// MI455X (gfx1250) — compile-verified
//
#include <hip/hip_runtime.h>
#include <cstdint>

#define SEQLEN 25
#define BS 256
#define DF 16384
#define NF 8
#define FRAME 2048
#define LAMF 0.1f
#define INVLAM 10.0f
#define WAVES 4
#define KC 64
#define LSTR 68  // 64 + 4 pad: 4-bank stagger per row -> conflict-free b64 gathers
#define NCHUNK (FRAME / KC)

typedef float v2f __attribute__((ext_vector_type(2)));
typedef float v8f __attribute__((ext_vector_type(8)));

// CDNA5 async global->LDS copy (ASYNCcnt-tracked, no VGPR round trip).
// saddr form: per-lane 32-bit voffset + uniform 64-bit SGPR base.
__device__ __forceinline__ void async_copy_b128(uint32_t lds_byte_addr,
                                                const void* sbase,
                                                uint32_t gvoff_bytes) {
  asm volatile("global_load_async_to_lds_b128 %0, %1, %2"
               :
               : "v"(lds_byte_addr), "v"(gvoff_bytes), "s"(sbase)
               : "memory");
}
__device__ __forceinline__ void wait_async0() {
  asm volatile("s_wait_asynccnt 0x0" ::: "memory");
}
// Generic LDS pointer -> 32-bit LDS byte address (aperture lives in high bits).
__device__ __forceinline__ uint32_t lds_addr(const void* p) {
  return (uint32_t)(uintptr_t)p;
}

// ---------------------------------------------------------------------------
// Kernel 1: per-(b,frame) inverse L2 norm of target frames -> d_ws
// ---------------------------------------------------------------------------
__global__ __launch_bounds__(256) void tgt_invnorm_kernel(
    const float* __restrict__ tgt, float* __restrict__ invT) {
  const int b = blockIdx.x;
  const int w = threadIdx.x >> 5;
  const int lane = threadIdx.x & 31;
  const size_t base = (size_t)b * DF + (size_t)w * FRAME;
  float ss = 0.f;
#pragma unroll
  for (int it = 0; it < 16; ++it) {
    const float4 v =
        *reinterpret_cast<const float4*>(tgt + base + (size_t)(it * 32 + lane) * 4);
    ss += v.x * v.x + v.y * v.y + v.z * v.z + v.w * v.w;
  }
#pragma unroll
  for (int off = 16; off > 0; off >>= 1) ss += __shfl_xor(ss, off, 32);
  if (lane == 0) invT[b * NF + w] = 1.0f / fmaxf(sqrtf(ss), 1e-12f);
}

__device__ __forceinline__ float lse2(float a, float b) {
  const float m = fmaxf(a, b);
  const float n = fminf(a, b);
  return m + LAMF * log1pf(expf((n - m) * INVLAM));
}

// ---------------------------------------------------------------------------
// Kernel 2: fused cosine-matrix (fp32 WMMA) + OTAM soft-DP.
// Double-buffered async global->LDS staging overlaps HBM with WMMA compute.
// ---------------------------------------------------------------------------
__global__ __launch_bounds__(128) void otam_kernel(
    const float* __restrict__ sup, const float* __restrict__ tgt,
    const float* __restrict__ invT, float* __restrict__ out) {
  __shared__ float sA[2][WAVES][16 * LSTR];  // support chunks per wave
  __shared__ float sB[2][16 * LSTR];         // target chunks (rows 8..15 = 0)
  __shared__ float sD[WAVES][16][8];         // scaled 2x(8x8) cosine matrices

  const int b = blockIdx.x >> 2;
  const int blk = blockIdx.x & 3;
  const int w = threadIdx.x >> 5;
  const int lane = threadIdx.x & 31;
  const int t = blk * WAVES + w;  // wave task 0..15 (valid: t < 13)
  const int s0 = t * 2;
  const int s1 = t * 2 + 1;
  const int sl0 = min(s0, SEQLEN - 1);
  const int sl1 = min(s1, SEQLEN - 1);

  // zero B rows 8..15 in both buffers (B cols 8..15 contribute 0)
  for (int i = threadIdx.x; i < 8 * LSTR; i += 128) {
    sB[0][8 * LSTR + i] = 0.f;
    sB[1][8 * LSTR + i] = 0.f;
  }

  const float* supRow0 = sup + (size_t)(sl0 * BS + b) * DF;
  const float* supRow1 = sup + (size_t)(sl1 * BS + b) * DF;
  const float* tgtB = tgt + (size_t)b * DF;

  // Per-lane offsets for 2-rows-per-instruction async staging:
  // lanes 0..15 -> row r (16B each), lanes 16..31 -> row r+1 (+8KB / +LSTR row).
  const uint32_t loff = (uint32_t)((lane & 15) * 16 + (lane >> 4) * (LSTR * 4));
  const uint32_t goff = (uint32_t)((lane & 15) * 16 + (lane >> 4) * (FRAME * 4));

  auto stage = [&](int buf, int c) {
    const int kbase = c * KC;
#pragma unroll
    for (int p = 0; p < 8; ++p) {
      const int r0 = 2 * p;  // row pairs never cross the s0/s1 boundary
      const float* gb =
          (r0 < 8 ? supRow0 : supRow1) + (size_t)(r0 & 7) * FRAME + kbase;
      async_copy_b128(lds_addr(&sA[buf][w][r0 * LSTR]) + loff, gb, goff);
    }
    {  // target frames 2w, 2w+1 staged by wave w (L2-resident stream)
      const float* gb = tgtB + (size_t)(2 * w) * FRAME + kbase;
      async_copy_b128(lds_addr(&sB[buf][(2 * w) * LSTR]) + loff, gb, goff);
    }
  };

  const int rowA = lane & 15;         // M (A) / N (B) fragment index
  const int koff = (lane >> 4) << 1;  // K sub-offset per lane half
  v8f acc = {};
  float ss = 0.f;

  stage(0, 0);
  wait_async0();
  __syncthreads();

  for (int c = 0; c < NCHUNK; ++c) {
    const int cur = c & 1;
    if (c + 1 < NCHUNK) stage(cur ^ 1, c + 1);  // prefetch next chunk (async)
#pragma unroll
    for (int kk = 0; kk < KC; kk += 4) {
      const v2f a =
          *reinterpret_cast<const v2f*>(&sA[cur][w][rowA * LSTR + kk + koff]);
      const v2f bb =
          *reinterpret_cast<const v2f*>(&sB[cur][rowA * LSTR + kk + koff]);
      ss += a.x * a.x + a.y * a.y;
      acc = __builtin_amdgcn_wmma_f32_16x16x4_f32(false, a, false, bb, (short)0,
                                                  acc, false, false);
    }
    if (c + 1 < NCHUNK) {
      wait_async0();    // own async writes (incl. shared target rows) landed
      __syncthreads();  // other waves' target rows landed / reads of cur done
    }
  }

  // Support frame inverse norms: lanes l and l+16 jointly cover row (l&15).
  ss += __shfl_xor(ss, 16, 32);
  const float invS = 1.0f / fmaxf(sqrtf(ss), 1e-12f);  // row (lane&15)
  const int Ncol = lane & 15;
  const float invTc = (Ncol < 8) ? invT[b * NF + Ncol] : 0.f;
  const int mbase = (lane & 16) >> 1;  // 0 (lanes 0-15) or 8 (lanes 16-31)
#pragma unroll
  for (int r = 0; r < 8; ++r) {
    const float inS = __shfl(invS, mbase + r, 32);  // invS of row M=mbase+r
    const float d = acc[r] * inS * invTc;
    if (Ncol < 8) sD[w][mbase + r][Ncol] = d;
  }
  __syncthreads();

  // 8x8 soft-alignment DP: lane0 s0-fwd, lane1 s0-T, lane2 s1-fwd, lane3 s1-T.
  float result = 0.f;
  if (lane < 4) {
    const int tr = lane & 1;
    const int rbase = ((lane >> 1) & 1) * 8;
    float prev[8];
#pragma unroll
    for (int i = 0; i < 8; ++i)
      prev[i] = tr ? sD[w][rbase + 0][i] : sD[w][rbase + i][0];
    for (int j = 1; j < 8; ++j) {
      const float n0 = prev[0] + (tr ? sD[w][rbase + j][0] : sD[w][rbase + 0][j]);
#pragma unroll
      for (int i = 7; i >= 1; --i) {
        const float dij = tr ? sD[w][rbase + j][i] : sD[w][rbase + i][j];
        prev[i] = lse2(prev[i - 1], prev[i]) + dij;
      }
      prev[0] = n0;
    }
    float m = prev[0];
#pragma unroll
    for (int i = 1; i < 8; ++i) m = fmaxf(m, prev[i]);
    float sum = 0.f;
#pragma unroll
    for (int i = 0; i < 8; ++i) sum += expf((prev[i] - m) * INVLAM);
    result = m + LAMF * logf(sum);
  }
  const float other = __shfl_down(result, 1, 32);
  if (lane == 0 && s0 < SEQLEN) out[s0 * BS + b] = 0.5f * (result + other);
  if (lane == 2 && s1 < SEQLEN) out[s1 * BS + b] = 0.5f * (result + other);
}

// ---------------------------------------------------------------------------
extern "C" void kernel_launch(void* const* d_in, const int* in_sizes, int n_in,
                              void* d_out, int out_size, void* d_ws, size_t ws_size,
                              hipStream_t stream) {
  const float* sup = (const float*)d_in[0];  // [25,256,16384] f32
  const float* tgt = (const float*)d_in[1];  // [256,16384]    f32
  float* out = (float*)d_out;                // [25,256]       f32
  float* invT = (float*)d_ws;                // 2048 f32 scratch

  tgt_invnorm_kernel<<<BS, 256, 0, stream>>>(tgt, invT);
  otam_kernel<<<BS * 4, 128, 0, stream>>>(sup, tgt, invT, out);
}